// YOLOloss_61804579389965
// MI455X (gfx1250) — compile-verified
//
#include <hip/hip_runtime.h>
#include <math.h>

#define SDIM 7
#define NCLS 20
#define NCH_P 30   // C + 10
#define NCH_T 25   // C + 5
#define LAMBDA_COORD 5.0f
#define LAMBDA_NOOBJ 0.5f
#define EPSV 1e-6f

#define TPB 256
#define CELLS_PER_BLOCK 256

typedef __attribute__((ext_vector_type(2))) float v2f;
typedef __attribute__((ext_vector_type(8))) float v8f;

#if __has_builtin(__builtin_amdgcn_global_load_async_to_lds_b128)
#define HAVE_ASYNC_LDS 1
#else
#define HAVE_ASYNC_LDS 0
#endif

#if HAVE_ASYNC_LDS
typedef int v4i __attribute__((ext_vector_type(4)));
typedef __attribute__((address_space(1))) v4i* gbl_v4i;
typedef __attribute__((address_space(3))) v4i* lds_v4i;
#endif

__device__ __forceinline__ void wait_async_all() {
#if HAVE_ASYNC_LDS
#if __has_builtin(__builtin_amdgcn_s_wait_asynccnt)
    __builtin_amdgcn_s_wait_asynccnt(0);
#else
    asm volatile("s_wait_asynccnt 0x0" ::: "memory");
#endif
#endif
}

__global__ void zero_out_kernel(float* out) { out[0] = 0.0f; }

__global__ __launch_bounds__(TPB) void yolo_loss_kernel(
    const float* __restrict__ pred, const float* __restrict__ targ,
    float* __restrict__ out, int n_cells, float inv_bs)
{
    __shared__ float lds_p[CELLS_PER_BLOCK * NCH_P]; // 7680 floats = 30720 B
    __shared__ float lds_t[CELLS_PER_BLOCK * NCH_T]; // 6400 floats = 25600 B

    const int tid  = threadIdx.x;
    const int base = blockIdx.x * CELLS_PER_BLOCK;
    const int cells = min(CELLS_PER_BLOCK, n_cells - base);

    // ---- Stage this block's chunk into LDS (coalesced b128; async DMA path on gfx1250) ----
    if (cells == CELLS_PER_BLOCK) {
        const float4* gp = (const float4*)(pred + (size_t)base * NCH_P); // 16B aligned
        const float4* gt = (const float4*)(targ + (size_t)base * NCH_T); // 16B aligned
        float4* lp = (float4*)lds_p;
        float4* lt = (float4*)lds_t;
#if HAVE_ASYNC_LDS
        // Direct global->LDS DMA: 16B/lane/instruction, tracked by ASYNCcnt.
        for (int i = tid; i < CELLS_PER_BLOCK * NCH_P / 4; i += TPB)
            __builtin_amdgcn_global_load_async_to_lds_b128(
                (gbl_v4i)(gp + i), (lds_v4i)(lp + i), 0, 0);
        for (int i = tid; i < CELLS_PER_BLOCK * NCH_T / 4; i += TPB)
            __builtin_amdgcn_global_load_async_to_lds_b128(
                (gbl_v4i)(gt + i), (lds_v4i)(lt + i), 0, 0);
        wait_async_all();
#else
        for (int i = tid; i < CELLS_PER_BLOCK * NCH_P / 4; i += TPB) lp[i] = gp[i];
        for (int i = tid; i < CELLS_PER_BLOCK * NCH_T / 4; i += TPB) lt[i] = gt[i];
#endif
    } else {
        for (int i = tid; i < cells * NCH_P; i += TPB)
            lds_p[i] = pred[(size_t)base * NCH_P + i];
        for (int i = tid; i < cells * NCH_T; i += TPB)
            lds_t[i] = targ[(size_t)base * NCH_T + i];
    }
    __syncthreads();

    // ---- Per-thread cell loss (stride-30/25 LDS reads: bank-conflict-free in wave32) ----
    float acc = 0.0f;
    if (tid < cells) {
        const float* p = lds_p + tid * NCH_P;
        const float* t = lds_t + tid * NCH_T;

        const float b1x = p[20], b1y = p[21], b1w = p[22], b1h = p[23], b1c = p[24];
        const float b2x = p[25], b2y = p[26], b2w = p[27], b2h = p[28], b2c = p[29];
        const float tx = t[20], ty = t[21], tw = t[22], th = t[23], tc = t[24];

        // Target box extents / area (shared by both IoUs)
        const float bx1 = tx - tw * 0.5f, bx2 = tx + tw * 0.5f;
        const float by1 = ty - th * 0.5f, by2 = ty + th * 0.5f;
        const float area_b = fabsf(tw * th);

        // IoU(box1, target)
        float ax1 = b1x - b1w * 0.5f, ax2 = b1x + b1w * 0.5f;
        float ay1 = b1y - b1h * 0.5f, ay2 = b1y + b1h * 0.5f;
        float iw = fmaxf(fminf(ax2, bx2) - fmaxf(ax1, bx1), 0.0f);
        float ih = fmaxf(fminf(ay2, by2) - fmaxf(ay1, by1), 0.0f);
        float inter = iw * ih;
        const float iou1 = inter / (fabsf(b1w * b1h) + area_b - inter + 1e-6f);

        // IoU(box2, target)
        ax1 = b2x - b2w * 0.5f; ax2 = b2x + b2w * 0.5f;
        ay1 = b2y - b2h * 0.5f; ay2 = b2y + b2h * 0.5f;
        iw = fmaxf(fminf(ax2, bx2) - fmaxf(ax1, bx1), 0.0f);
        ih = fmaxf(fminf(ay2, by2) - fmaxf(ay1, by1), 0.0f);
        inter = iw * ih;
        const float iou2 = inter / (fabsf(b2w * b2h) + area_b - inter + 1e-6f);

        const bool use1 = iou1 > iou2;
        const float rx = use1 ? b1x : b2x;
        const float ry = use1 ? b1y : b2y;
        const float rw = use1 ? b1w : b2w;
        const float rh = use1 ? b1h : b2h;
        const float rc = use1 ? b1c : b2c;
        const float other = use1 ? b2c : b1c;

        const float dx = rx - tx, dy = ry - ty;
        const float dsw = sqrtf(fmaxf(rw, EPSV)) - sqrtf(fmaxf(tw, EPSV));
        const float dsh = sqrtf(fmaxf(rh, EPSV)) - sqrtf(fmaxf(th, EPSV));
        const float coord = LAMBDA_COORD * (dx * dx + dy * dy + dsw * dsw + dsh * dsh);

        const float dc = rc - tc;
        const float obj_conf = dc * dc;

        float cls = 0.0f;
        #pragma unroll
        for (int i = 0; i < NCLS; ++i) {
            const float d = p[i] - t[i];
            cls += d * d;
        }

        const float obj = (t[20] == 1.0f) ? 1.0f : 0.0f;
        const float noobj_in  = LAMBDA_NOOBJ * other * other;
        const float noobj_out = LAMBDA_NOOBJ * (b1c * b1c + b2c * b2c);

        acc = obj * (coord + obj_conf + cls + noobj_in) + (1.0f - obj) * noobj_out;
    }
    __syncthreads();   // done reading staged data

    // ---- Block reduction of 256 partials via chained V_WMMA_F32_16X16X4_F32 ----
    lds_p[tid] = acc;  // reuse staging LDS for partials
    __syncthreads();

    if (tid < 32) {    // wave 0 only: EXEC all-1s within the wave
        const int m  = tid & 15;
        const int hi = tid >> 4;
        v2f b; b[0] = 1.0f; b[1] = 1.0f;   // B = ones(4x16): D[i][j] = sum_k A[i][k] + C
        v8f d = {};
        #pragma unroll
        for (int q = 0; q < 4; ++q) {
            // A tile q: A[m][k] = partial[q*64 + m*4 + k]; lanes 0-15 hold K=0,1; 16-31 hold K=2,3
            const float* src = lds_p + q * 64 + m * 4 + hi * 2;
            v2f a; a[0] = src[0]; a[1] = src[1];
            d = __builtin_amdgcn_wmma_f32_16x16x4_f32(
                    false, a, false, b, (short)0, d, false, false);
        }
        // lanes 0-15: d[v] = rowsum s_v (v=0..7); lanes 16-31: s_{v+8}
        float local = d[0] + d[1] + d[2] + d[3] + d[4] + d[5] + d[6] + d[7];
        local += __shfl_xor(local, 16, 32);   // S_lo + S_hi = sum of all 256 partials
        if (tid == 0) atomicAdd(out, local * inv_bs);
    }
}

extern "C" void kernel_launch(void* const* d_in, const int* in_sizes, int n_in,
                              void* d_out, int out_size, void* d_ws, size_t ws_size,
                              hipStream_t stream) {
    const float* pred = (const float*)d_in[0];
    const float* targ = (const float*)d_in[1];
    float* out = (float*)d_out;

    const int n_cells = in_sizes[0] / NCH_P;          // BATCH * S * S = 802816
    const int batch   = n_cells / (SDIM * SDIM);      // 16384
    const float inv_bs = 1.0f / (float)batch;

    zero_out_kernel<<<1, 1, 0, stream>>>(out);

    const int blocks = (n_cells + CELLS_PER_BLOCK - 1) / CELLS_PER_BLOCK; // 3136
    yolo_loss_kernel<<<blocks, TPB, 0, stream>>>(pred, targ, out, n_cells, inv_bs);
}